// UltraGCN_33560874451737
// MI455X (gfx1250) — compile-verified
//
#include <hip/hip_runtime.h>
#include <hip/hip_bf16.h>
#include <cstddef>

// ---------------- problem constants ----------------
#define NU      500000
#define NI      500000
#define ND      64
#define NK      32
#define NB      8192
#define NNEG    128
#define W1C     1e-6f
#define W2C     1.0f
#define W3C     1e-6f
#define W4C     1.0f
#define NEGW    200.0f
#define GAMMAC  1e-4f
#define LAMBDAC 2.5f

typedef __attribute__((ext_vector_type(2))) float v2f;
typedef __attribute__((ext_vector_type(8))) float v8f;
typedef int v4i __attribute__((vector_size(4 * sizeof(int))));

#define AS_GLOBAL __attribute__((address_space(1)))
#define AS_LDS    __attribute__((address_space(3)))

#if __has_builtin(__builtin_amdgcn_global_load_async_to_lds_b128)
#define HAVE_ASYNC_LDS 1
#else
#define HAVE_ASYNC_LDS 0
#endif

__device__ __forceinline__ float softplusf(float x) {
    // stable softplus: max(x,0) + log1p(exp(-|x|))
    return fmaxf(x, 0.0f) + log1pf(__expf(-fabsf(x)));
}

__device__ __forceinline__ float waveReduceSum(float v) {
    #pragma unroll
    for (int m = 16; m >= 1; m >>= 1) v += __shfl_xor(v, m, 32);
    return v;
}

// ws float accumulators: [0]=lossL_sum [1]=lossI_raw_sum [2]=mse_sum [3]=norm_sumsq
__global__ void init_acc_kernel(float* acc) {
    if (threadIdx.x < 4) acc[threadIdx.x] = 0.0f;
}

// ---------------- norm: sum of squares over all params (streaming, fp32) ----
__global__ void norm_kernel(const float* __restrict__ ue, const float* __restrict__ ie,
                            const float* __restrict__ w0, const float* __restrict__ b0,
                            const float* __restrict__ w1, const float* __restrict__ b1,
                            const float* __restrict__ w2, const float* __restrict__ b2,
                            const float* __restrict__ w3, const float* __restrict__ b3,
                            float* __restrict__ acc) {
    const size_t tid    = (size_t)blockIdx.x * blockDim.x + threadIdx.x;
    const size_t stride = (size_t)gridDim.x * blockDim.x;
    float s = 0.0f;

    const size_t n4 = (size_t)NU * ND / 4;   // 8M float4 per table
    const float4* a4 = (const float4*)ue;
    const float4* c4 = (const float4*)ie;
    for (size_t i = tid; i < n4; i += stride) {
        float4 x = a4[i]; s += x.x*x.x + x.y*x.y + x.z*x.z + x.w*x.w;
        float4 y = c4[i]; s += y.x*y.x + y.y*y.y + y.z*y.z + y.w*y.w;
    }
    for (size_t i = tid; i < (size_t)ND*ND; i += stride) {      // mse_u_w, mse_i_w
        float x = w0[i]; s += x*x;
        float y = w1[i]; s += y*y;
    }
    for (size_t i = tid; i < (size_t)ND*2*ND; i += stride) {    // fc1_w1
        float x = w2[i]; s += x*x;
    }
    for (size_t i = tid; i < (size_t)ND; i += stride) {         // biases + fc1_w2
        s += b0[i]*b0[i] + b1[i]*b1[i] + b2[i]*b2[i] + w3[i]*w3[i];
    }
    if (tid == 0) s += b3[0]*b3[0];                              // fc1_b2

    s = waveReduceSum(s);
    __shared__ float sh[8];
    const int lane = threadIdx.x & 31, wv = threadIdx.x >> 5;
    if (lane == 0) sh[wv] = s;
    __syncthreads();
    if (threadIdx.x == 0) {
        float t = 0.0f;
        #pragma unroll
        for (int w = 0; w < 8; ++w) t += sh[w];
        atomicAdd(&acc[3], t);
    }
}

// ---------------- loss_L + loss_I: one block (8 waves) per batch row --------
__global__ void lossLI_kernel(const int* __restrict__ users, const int* __restrict__ pos_items,
                              const int* __restrict__ neg_items,
                              const float* __restrict__ user_emb, const float* __restrict__ item_emb,
                              const float* __restrict__ beta_uD, const float* __restrict__ beta_iD,
                              const int* __restrict__ ii_nbr, const float* __restrict__ ii_con,
                              float* __restrict__ acc) {
    const int b    = blockIdx.x;
    const int tid  = threadIdx.x;
    const int lane = tid & 31;
    const int wv   = tid >> 5;                 // 8 waves

    __shared__ float sNeg[8], sI[8], sPos;

    const int   u  = users[b];
    const int   pi = pos_items[b];
    const float bu = beta_uD[u];
    const float2 uv = *(const float2*)(user_emb + (size_t)u * ND + lane * 2);

    float negAcc = 0.0f, iAcc = 0.0f;

    // positive score (wave 0): fully coalesced 256B row, wave reduce
    if (wv == 0) {
        const float2 pv = *(const float2*)(item_emb + (size_t)pi * ND + lane * 2);
        float s = waveReduceSum(uv.x * pv.x + uv.y * pv.y);
        if (lane == 0) {
            float posw = W1C + W2C * bu * beta_iD[pi];
            sPos = posw * softplusf(-s);
        }
    }

    // 128 negatives, 16 per wave; prefetch next gathered row
    for (int n = wv; n < NNEG; n += 8) {
        const int idx = neg_items[b * NNEG + n];
        if (n + 8 < NNEG) {
            const int idx2 = neg_items[b * NNEG + n + 8];
            __builtin_prefetch(item_emb + (size_t)idx2 * ND + lane * 2, 0, 0);
        }
        const float2 v = *(const float2*)(item_emb + (size_t)idx * ND + lane * 2);
        float s = waveReduceSum(uv.x * v.x + uv.y * v.y);
        if (lane == 0) {
            float nw = W3C + W4C * bu * beta_iD[idx];
            negAcc += nw * softplusf(s);
        }
    }

    // 32 item-item neighbors of pos item, 4 per wave
    for (int k = wv; k < NK; k += 8) {
        const int idx = ii_nbr[(size_t)pi * NK + k];
        const float2 v = *(const float2*)(item_emb + (size_t)idx * ND + lane * 2);
        float s = waveReduceSum(uv.x * v.x + uv.y * v.y);
        if (lane == 0) iAcc += ii_con[(size_t)pi * NK + k] * softplusf(-s);
    }

    if (lane == 0) { sNeg[wv] = negAcc; sI[wv] = iAcc; }
    __syncthreads();
    if (tid == 0) {
        float ns = 0.0f, is = 0.0f;
        #pragma unroll
        for (int w = 0; w < 8; ++w) { ns += sNeg[w]; is += sI[w]; }
        atomicAdd(&acc[0], sPos + NEGW * (ns * (1.0f / NNEG)));
        atomicAdd(&acc[1], is);
    }
}

// ---------------- mse path on WMMA: feat = emb @ W^T + b, pred dot ----------
// Block = 128 threads (4 waves). Each wave owns 16 batch rows and computes
// both 16x64 feature tiles with V_WMMA_F32_16X16X4_F32 (16 K-steps, 4 N-tiles).
// Weights staged into LDS with CDNA5 async global->LDS (no VGPR round trip,
// tracked on ASYNCcnt) when the toolchain declares the builtin.
__global__ void __launch_bounds__(128)
mse_wmma_kernel(const int* __restrict__ users, const int* __restrict__ pos_items,
                const float* __restrict__ user_emb, const float* __restrict__ item_emb,
                const float* __restrict__ mse_u_w, const float* __restrict__ mse_u_b,
                const float* __restrict__ mse_i_w, const float* __restrict__ mse_i_b,
                const float* __restrict__ rpkms, float* __restrict__ acc) {
    __shared__ __align__(16) float lds_w[2 * ND * ND];  // 32 KB: [0..4095]=u_w, [4096..]=i_w

#if HAVE_ASYNC_LDS
    // async stage both 64x64 weight matrices: 16B per lane-op, global->LDS direct
    for (int i = threadIdx.x * 4; i < ND * ND; i += 128 * 4) {
        __builtin_amdgcn_global_load_async_to_lds_b128(
            (AS_GLOBAL v4i*)(mse_u_w + i), (AS_LDS v4i*)(lds_w + i), 0, 0);
        __builtin_amdgcn_global_load_async_to_lds_b128(
            (AS_GLOBAL v4i*)(mse_i_w + i), (AS_LDS v4i*)(lds_w + ND * ND + i), 0, 0);
    }
#if __has_builtin(__builtin_amdgcn_s_wait_asynccnt)
    __builtin_amdgcn_s_wait_asynccnt(0);
#else
    asm volatile("s_wait_asynccnt 0x0" ::: "memory");
#endif
#else
    for (int i = threadIdx.x; i < ND * ND; i += 128) {
        lds_w[i]           = mse_u_w[i];
        lds_w[ND * ND + i] = mse_i_w[i];
    }
#endif
    __syncthreads();

    const int lane = threadIdx.x & 31;
    const int wv   = threadIdx.x >> 5;
    const int m    = lane & 15;              // A-matrix row / B,C column (N) index
    const int half = lane >> 4;
    const int koff = half ? 2 : 0;           // A/B fragment K offset per ISA layout
    const int rowbase = (blockIdx.x * 4 + wv) * 16;

    const float* __restrict__ urow = user_emb + (size_t)users[rowbase + m]     * ND;
    const float* __restrict__ irow = item_emb + (size_t)pos_items[rowbase + m] * ND;

    v8f cu[4] = {v8f{}, v8f{}, v8f{}, v8f{}};
    v8f ci[4] = {v8f{}, v8f{}, v8f{}, v8f{}};

    #pragma unroll 4
    for (int kk = 0; kk < 16; ++kk) {
        const int k = kk * 4 + koff;
        v2f au; au.x = urow[k]; au.y = urow[k + 1];
        v2f ai; ai.x = irow[k]; ai.y = irow[k + 1];
        #pragma unroll
        for (int nt = 0; nt < 4; ++nt) {
            const int col = nt * 16 + m;     // output feature index j; B[k][j] = W[j][k]
            v2f bu_; bu_.x = lds_w[col * ND + k];           bu_.y = lds_w[col * ND + k + 1];
            v2f bi_; bi_.x = lds_w[ND*ND + col * ND + k];   bi_.y = lds_w[ND*ND + col * ND + k + 1];
            cu[nt] = __builtin_amdgcn_wmma_f32_16x16x4_f32(
                false, au, false, bu_, (short)0, cu[nt], false, false);
            ci[nt] = __builtin_amdgcn_wmma_f32_16x16x4_f32(
                false, ai, false, bi_, (short)0, ci[nt], false, false);
        }
    }

    // bias add (column j = nt*16 + m for every C VGPR)
    #pragma unroll
    for (int nt = 0; nt < 4; ++nt) {
        const float bju = mse_u_b[nt * 16 + m];
        const float bji = mse_i_b[nt * 16 + m];
        #pragma unroll
        for (int r = 0; r < 8; ++r) { cu[nt][r] += bju; ci[nt][r] += bji; }
    }

    // pred[b] = dot(feat_u[b], feat_i[b]); C layout: VGPR r holds row r + 8*half
    float local = 0.0f;
    #pragma unroll
    for (int r = 0; r < 8; ++r) {
        float s = 0.0f;
        #pragma unroll
        for (int nt = 0; nt < 4; ++nt) s += cu[nt][r] * ci[nt][r];
        #pragma unroll
        for (int msk = 8; msk >= 1; msk >>= 1) s += __shfl_xor(s, msk, 32); // 16-lane group
        if (m == r) {                      // one lane per row picks the result
            const float e = s - rpkms[rowbase + r + 8 * half];
            local = e * e;
        }
    }
    local = waveReduceSum(local);          // 16 rows' squared errors per wave
    if (lane == 0) atomicAdd(&acc[2], local);
}

// ---------------- finalize ----------------
__global__ void finalize_kernel(const float* __restrict__ acc, float* __restrict__ out) {
    const float lossL = acc[0] * (1.0f / NB);
    const float lossI = LAMBDAC * acc[1] * (1.0f / ((float)NB * NK));
    const float mse   = acc[2] * (1.0f / NB);
    const float norm  = GAMMAC * 0.5f * acc[3];
    out[0] = lossL + norm + lossI + mse;
    out[1] = lossL;
    out[2] = lossI;
    out[3] = 0.0f;
    out[4] = mse;
    out[5] = norm;
}

extern "C" void kernel_launch(void* const* d_in, const int* in_sizes, int n_in,
                              void* d_out, int out_size, void* d_ws, size_t ws_size,
                              hipStream_t stream) {
    const int*   users      = (const int*)  d_in[0];
    const int*   pos_items  = (const int*)  d_in[1];
    const int*   neg_items  = (const int*)  d_in[2];
    const float* rpkms      = (const float*)d_in[3];
    const float* user_emb   = (const float*)d_in[4];
    const float* item_emb   = (const float*)d_in[5];
    const float* beta_uD    = (const float*)d_in[6];
    const float* beta_iD    = (const float*)d_in[7];
    const int*   ii_nbr     = (const int*)  d_in[8];
    const float* ii_con     = (const float*)d_in[9];
    const float* mse_u_w    = (const float*)d_in[10];
    const float* mse_u_b    = (const float*)d_in[11];
    const float* mse_i_w    = (const float*)d_in[12];
    const float* mse_i_b    = (const float*)d_in[13];
    const float* fc1_w1     = (const float*)d_in[14];
    const float* fc1_b1     = (const float*)d_in[15];
    const float* fc1_w2     = (const float*)d_in[16];
    const float* fc1_b2     = (const float*)d_in[17];
    float* out = (float*)d_out;
    float* acc = (float*)d_ws;

    init_acc_kernel<<<1, 32, 0, stream>>>(acc);

    norm_kernel<<<2048, 256, 0, stream>>>(user_emb, item_emb,
                                          mse_u_w, mse_u_b, mse_i_w, mse_i_b,
                                          fc1_w1, fc1_b1, fc1_w2, fc1_b2, acc);

    lossLI_kernel<<<NB, 256, 0, stream>>>(users, pos_items, neg_items,
                                          user_emb, item_emb, beta_uD, beta_iD,
                                          ii_nbr, ii_con, acc);

    mse_wmma_kernel<<<NB / 64, 128, 0, stream>>>(users, pos_items, user_emb, item_emb,
                                                 mse_u_w, mse_u_b, mse_i_w, mse_i_b,
                                                 rpkms, acc);

    finalize_kernel<<<1, 1, 0, stream>>>(acc, out);
}